// ChamferDistance_23433341567534
// MI455X (gfx1250) — compile-verified
//
#include <hip/hip_runtime.h>

typedef __attribute__((ext_vector_type(2))) float v2f;
typedef __attribute__((ext_vector_type(8))) float v8f;
typedef int v4i __attribute__((vector_size(16)));   // matches async-LDS builtin param

#define AS1 __attribute__((address_space(1)))
#define AS3 __attribute__((address_space(3)))

constexpr int C         = 128;     // feature dim
constexpr int NPTS      = 8192;    // points per cloud
constexpr int BATCH     = 4;
constexpr int TILE_ROWS = 128;     // A-rows per block (8 waves x 16 rows)
constexpr int WAVES     = 8;
constexpr int KCHUNKS   = C / 4;   // 32 K-chunks of 4 for 16x16x4 WMMA
constexpr int MT        = 32;      // B-points per main-loop iteration
constexpr int NORMN     = BATCH * NPTS;   // norms per input array

#if __has_builtin(__builtin_amdgcn_global_load_async_to_lds_b128)
#define HAVE_ASYNC_LDS 1
#else
#define HAVE_ASYNC_LDS 0
#endif

__device__ __forceinline__ void wait_asynccnt0() {
#if __has_builtin(__builtin_amdgcn_s_wait_asynccnt)
    __builtin_amdgcn_s_wait_asynccnt(0);
#else
    asm volatile("s_wait_asynccnt 0x0" ::: "memory");
#endif
}

__device__ __forceinline__ void async_copy16(const float* g, float* l) {
#if HAVE_ASYNC_LDS
    __builtin_amdgcn_global_load_async_to_lds_b128(
        (AS1 v4i*)g, (AS3 v4i*)l, 0, 0);
#endif
}

// ---- Precompute squared row norms of f (slot 0) and f_ (slot 1) into ws.
__global__ __launch_bounds__(256)
void row_norms(const float* __restrict__ f, const float* __restrict__ f_,
               float* __restrict__ norms)
{
    const float* src = blockIdx.y ? f_ : f;
    float* dst = norms + (size_t)blockIdx.y * NORMN;
    const int row = blockIdx.x * 16 + (threadIdx.x >> 4);
    const int col = (threadIdx.x & 15) * 8;
    const float4* p = (const float4*)(src + (size_t)row * C + col);
    float4 a = p[0], b = p[1];
    float s = a.x * a.x + a.y * a.y + a.z * a.z + a.w * a.w
            + b.x * b.x + b.y * b.y + b.z * b.z + b.w * b.w;
    s += __shfl_xor(s, 1);
    s += __shfl_xor(s, 2);
    s += __shfl_xor(s, 4);
    s += __shfl_xor(s, 8);
    if ((threadIdx.x & 15) == 0) dst[row] = s;
}

// One-sided chamfer with f32 WMMA + double-buffered async global->LDS pipeline.
// gridDim = (NPTS/TILE_ROWS, BATCH, 2); z==0: A=f,B=f_ ; z==1: swapped.
__global__ __launch_bounds__(256)
void chamfer_onesided_wmma(const float* __restrict__ F,
                           const float* __restrict__ F2,
                           const float* __restrict__ norms,
                           float* __restrict__ partials)
{
    __shared__ __align__(16) float sB[2][MT * C];   // 2 x 16 KB B tiles
    __shared__ float sWave[WAVES];

    const int side = blockIdx.z;
    const float* A = side ? F2 : F;
    const float* B = side ? F  : F2;
    const int b    = blockIdx.y;
    const int tid  = threadIdx.x;
    const int lane = tid & 31;
    const int wave = tid >> 5;

    const float* Abase = A + (size_t)b * NPTS * C;
    const float* Bbase = B + (size_t)b * NPTS * C;
    // norms of the B-side array: side==0 -> f_ (slot 1), side==1 -> f (slot 0)
    const float* nB = norms + (side ? 0 : (size_t)NORMN) + (size_t)b * NPTS;

    // ---- This wave's 16x128 A tile in WMMA A-frag layout (registers).
    // 16x4 f32 A layout: lanes 0-15 -> (M=lane, K=0,1), lanes 16-31 -> (M=lane-16, K=2,3)
    const int arow = blockIdx.x * TILE_ROWS + wave * 16 + (lane & 15);
    const int koff = (lane >> 4) * 2;               // 0 or 2
    const float* aptr = Abase + (size_t)arow * C + koff;

    v2f af[KCHUNKS];
    float apart = 0.f;                              // this lane's half of |a_row|^2
    #pragma unroll
    for (int c = 0; c < KCHUNKS; ++c) {
        af[c] = *(const v2f*)(aptr + 4 * c);
        apart += af[c].x * af[c].x + af[c].y * af[c].y;
    }

    float vmin[8];
    #pragma unroll
    for (int r = 0; r < 8; ++r) vmin[r] = 3.0e38f;

    // Per-thread slice of the 32x128 tile copy: 4x b128 = 64 bytes
    const int crow = tid >> 3;                      // 0..31
    const int ccol = (tid & 7) * 16;                // 0..112

    constexpr int nIter = NPTS / MT;                // 256

#if HAVE_ASYNC_LDS
    // Prologue: async-load tile 0 into buffer 0.
    {
        const float* g = Bbase + (size_t)crow * C + ccol;
        float* l = &sB[0][crow * C + ccol];
        #pragma unroll
        for (int q = 0; q < 4; ++q)
            async_copy16(g + 4 * q, l + 4 * q);
    }
    wait_asynccnt0();
    __syncthreads();
#endif

    for (int mt = 0; mt < nIter; ++mt) {
#if HAVE_ASYNC_LDS
        const int cur = mt & 1;
        // Prefetch next tile into the other buffer (overlaps with WMMA below).
        if (mt + 1 < nIter) {
            const float* g = Bbase + (size_t)((mt + 1) * MT + crow) * C + ccol;
            float* l = &sB[cur ^ 1][crow * C + ccol];
            #pragma unroll
            for (int q = 0; q < 4; ++q)
                async_copy16(g + 4 * q, l + 4 * q);
        }
#else
        const int cur = 0;
        __syncthreads();
        {
            const float4* g = (const float4*)(Bbase + (size_t)(mt * MT + crow) * C + ccol);
            float4* l = (float4*)(&sB[0][crow * C + ccol]);
            float4 x0 = g[0], x1 = g[1], x2 = g[2], x3 = g[3];
            l[0] = x0; l[1] = x1; l[2] = x2; l[3] = x3;
        }
        __syncthreads();
#endif
        // Precomputed |b|^2 for this tile's 32 columns (uniform, L2-resident).
        const float bn0 = nB[mt * MT + (lane & 15)];
        const float bn1 = nB[mt * MT + 16 + (lane & 15)];

        // ---- Two 16x16 dot tiles via 64x V_WMMA_F32_16X16X4_F32 (4 acc chains).
        // B-frag (4x16) layout: lane l holds Btile[l&15][4c + koff + {0,1}]
        const float* bp0 = &sB[cur][(lane & 15) * C + koff];
        const float* bp1 = bp0 + 16 * C;
        v8f a0 = {}, a1 = {}, a2 = {}, a3 = {};
        #pragma unroll
        for (int c = 0; c < KCHUNKS; c += 2) {
            v2f x0 = *(const v2f*)(bp0 + 4 * c);
            v2f x1 = *(const v2f*)(bp0 + 4 * (c + 1));
            v2f y0 = *(const v2f*)(bp1 + 4 * c);
            v2f y1 = *(const v2f*)(bp1 + 4 * (c + 1));
            a0 = __builtin_amdgcn_wmma_f32_16x16x4_f32(
                false, af[c],     false, x0, (short)0, a0, false, false);
            a1 = __builtin_amdgcn_wmma_f32_16x16x4_f32(
                false, af[c + 1], false, x1, (short)0, a1, false, false);
            a2 = __builtin_amdgcn_wmma_f32_16x16x4_f32(
                false, af[c],     false, y0, (short)0, a2, false, false);
            a3 = __builtin_amdgcn_wmma_f32_16x16x4_f32(
                false, af[c + 1], false, y1, (short)0, a3, false, false);
        }

        // C/D layout: VGPR r, lane l -> row M = r + 8*(l>>4), col N = l&15
        #pragma unroll
        for (int r = 0; r < 8; ++r) {
            vmin[r] = fminf(vmin[r], fmaf(-2.f, a0[r] + a1[r], bn0));
            vmin[r] = fminf(vmin[r], fmaf(-2.f, a2[r] + a3[r], bn1));
        }

#if HAVE_ASYNC_LDS
        // Next tile must be resident and everyone done with `cur` before reuse.
        wait_asynccnt0();
        __syncthreads();
#endif
    }

    // ---- Reduce mins over the 16 columns held across each 16-lane half.
    #pragma unroll
    for (int r = 0; r < 8; ++r) {
        vmin[r] = fminf(vmin[r], __shfl_xor(vmin[r], 1));
        vmin[r] = fminf(vmin[r], __shfl_xor(vmin[r], 2));
        vmin[r] = fminf(vmin[r], __shfl_xor(vmin[r], 4));
        vmin[r] = fminf(vmin[r], __shfl_xor(vmin[r], 8));
    }
    float s = vmin[0] + vmin[1] + vmin[2] + vmin[3]
            + vmin[4] + vmin[5] + vmin[6] + vmin[7];
    s += __shfl_xor(s, 16);                         // all 16 rows of this wave

    float ap = apart;                               // sum(|a_row|^2) over wave's rows
    ap += __shfl_xor(ap, 1);
    ap += __shfl_xor(ap, 2);
    ap += __shfl_xor(ap, 4);
    ap += __shfl_xor(ap, 8);
    ap += __shfl_xor(ap, 16);

    if (lane == 0) sWave[wave] = s + ap;
    __syncthreads();
    if (tid == 0) {
        float t = 0.f;
        #pragma unroll
        for (int w = 0; w < WAVES; ++w) t += sWave[w];
        partials[(blockIdx.z * gridDim.y + blockIdx.y) * gridDim.x + blockIdx.x] = t;
    }
}

// Deterministic tree reduction of per-block partials -> scalar chamfer distance.
__global__ __launch_bounds__(256)
void chamfer_finalize(const float* __restrict__ partials, int n,
                      float* __restrict__ out)
{
    __shared__ float red[256];
    float s = 0.f;
    for (int i = threadIdx.x; i < n; i += 256) s += partials[i];
    red[threadIdx.x] = s;
    __syncthreads();
    for (int off = 128; off > 0; off >>= 1) {
        if (threadIdx.x < off) red[threadIdx.x] += red[threadIdx.x + off];
        __syncthreads();
    }
    if (threadIdx.x == 0)
        out[0] = red[0] / (float)(BATCH * NPTS);
}

extern "C" void kernel_launch(void* const* d_in, const int* in_sizes, int n_in,
                              void* d_out, int out_size, void* d_ws, size_t ws_size,
                              hipStream_t stream)
{
    const float* f  = (const float*)d_in[0];       // [4, 8192, 128] f32
    const float* f_ = (const float*)d_in[1];       // [4, 8192, 128] f32

    float* norms    = (float*)d_ws;                // 2 * 32768 floats
    float* partials = norms + 2 * (size_t)NORMN;   // 512 floats

    // 1) row norms of both clouds
    row_norms<<<dim3(NORMN / 16, 2), 256, 0, stream>>>(f, f_, norms);

    // 2) both chamfer directions
    dim3 grid(NPTS / TILE_ROWS, BATCH, 2);         // 64 x 4 x 2 = 512 blocks
    chamfer_onesided_wmma<<<grid, 256, 0, stream>>>(f, f_, norms, partials);

    // 3) deterministic scalar reduction
    const int nPartials = (NPTS / TILE_ROWS) * BATCH * 2;
    chamfer_finalize<<<1, 256, 0, stream>>>(partials, nPartials, (float*)d_out);
}